// AdptiveMLP_26929444946040
// MI455X (gfx1250) — compile-verified
//
#include <hip/hip_runtime.h>

typedef __attribute__((ext_vector_type(2))) float v2f;
typedef __attribute__((ext_vector_type(8))) float v8f;

#define G     10
#define CIN   3
#define H     64
#define COUT  3
#define MT    128      // batch rows per workgroup (8 waves x 16-row M-tiles)
#define K0AUG 40       // layer-0 augmented K (30 weight cols + 10 bias cols)
#define Z0S   42       // z0 LDS row stride (even -> 8B aligned; bank-spread)
#define COS   12       // co LDS row stride (10 padded to 12 so bias GEMM K%4==0)
#define HS    66       // hidden LDS row stride (even; bank-spread)
#define W0TS  42       // transposed W0aug row stride: w0t[n*W0TS + k]
#define B1TS  12       // transposed b1aug row stride: b1t[n*B1TS + k]

__device__ __forceinline__ v8f wmma_f32(v2f a, v2f b, v8f c) {
  // D = A(16x4,f32) * B(4x16,f32) + C(16x16,f32)
  return __builtin_amdgcn_wmma_f32_16x16x4_f32(
      /*neg_a=*/false, a, /*neg_b=*/false, b,
      /*c_mod=*/(short)0, c, /*reuse_a=*/false, /*reuse_b=*/false);
}

__global__ __launch_bounds__(256) void adaptive_mlp_kernel(
    const float* __restrict__ x,  const float* __restrict__ co,
    const float* __restrict__ W0, const float* __restrict__ W1,
    const float* __restrict__ W2, const float* __restrict__ b0,
    const float* __restrict__ b1, const float* __restrict__ b2,
    float* __restrict__ out)
{
  __shared__ float co_s[MT * COS];    //  6.0 KB  co rows, cols padded to 12 (zeros)
  __shared__ float w0t[H * W0TS];     // 10.5 KB  transposed [W0_flat(30x64); b0(10x64)]
  __shared__ float b1t[H * B1TS];     //  3.0 KB  transposed b1 padded to K=12
  __shared__ float z0_s[MT * Z0S];    // 21.0 KB  augmented layer-0 A operand
  __shared__ float h_s[MT * HS];      // 33.0 KB  hidden activations (h0 then h1)

  const int tid  = threadIdx.x;
  const int wave = tid >> 5;          // wave == M-tile index (8 x 16 rows)
  const int lane = tid & 31;
  const int hl   = lane >> 4;         // half-wave: selects K-pair / M-row group
  const int nm16 = lane & 15;         // M for A operand; N for B/C operands
  const int row0 = (int)blockIdx.x * MT;

  // ---- cooperative staging: co (padded), transposed W0||b0, transposed b1 ----
  for (int i = tid; i < MT * COS; i += 256) {
    const int r = i / COS, c = i - r * COS;
    co_s[i] = (c < G) ? co[(row0 + r) * G + c] : 0.0f;
  }
  for (int i = tid; i < K0AUG * H; i += 256) {   // read [k][n] coalesced, store transposed
    const int k = i >> 6, n = i & 63;
    const float v = (k < G * CIN) ? W0[k * H + n] : b0[(k - G * CIN) * H + n];
    w0t[n * W0TS + k] = v;
  }
  for (int i = tid; i < 12 * H; i += 256) {
    const int k = i >> 6, n = i & 63;
    b1t[n * B1TS + k] = (k < G) ? b1[k * H + n] : 0.0f;
  }
  __syncthreads();   // the ONLY barrier: everything below is private to each wave's rows

  // ---- per-wave build of augmented z0 rows: [co*x (30)] ++ [co (10)] ----
  const int r0 = wave * 16;
  for (int i = lane; i < 16 * K0AUG; i += 32) {
    const int rr = i / K0AUG, c = i - rr * K0AUG;
    const int r = r0 + rr;
    float v;
    if (c < G * CIN) {
      const int n = c / CIN, ii = c - n * CIN;
      v = co_s[r * COS + n] * x[(row0 + r) * CIN + ii];
    } else {
      v = co_s[r * COS + (c - G * CIN)];
    }
    z0_s[r * Z0S + c] = v;
  }

  const int ar = r0 + nm16;           // this lane's A-operand row

  // ---- layer 0: (16x40) @ (40x64) per wave; bias folded in K ----
  {
    v8f acc[4] = {};
    #pragma unroll
    for (int k0 = 0; k0 < K0AUG; k0 += 4) {
      const int k = k0 + 2 * hl;
      const v2f a = *(const v2f*)&z0_s[ar * Z0S + k];
      #pragma unroll
      for (int j = 0; j < 4; ++j) {
        const v2f b = *(const v2f*)&w0t[(j * 16 + nm16) * W0TS + k];
        acc[j] = wmma_f32(a, b, acc[j]);
      }
    }
    #pragma unroll
    for (int j = 0; j < 4; ++j)
      #pragma unroll
      for (int v = 0; v < 8; ++v)
        h_s[(r0 + v + 8 * hl) * HS + j * 16 + nm16] = fmaxf(acc[j][v], 0.0f);
  }

  // ---- layer 1: sum_g (co[:,g]*h0) @ W1[g]  +  co @ b1 (K padded to 12) ----
  {
    v8f acc[4] = {};
    for (int g = 0; g < G; ++g) {
      const float cg = co_s[ar * COS + g];
      const float* __restrict__ Wg = W1 + g * H * H;
      #pragma unroll
      for (int k0 = 0; k0 < H; k0 += 4) {
        const int k = k0 + 2 * hl;
        const v2f a = (*(const v2f*)&h_s[ar * HS + k]) * cg;
        #pragma unroll
        for (int j = 0; j < 4; ++j) {
          const int n = j * 16 + nm16;
          v2f b; b[0] = Wg[k * H + n]; b[1] = Wg[(k + 1) * H + n];
          acc[j] = wmma_f32(a, b, acc[j]);
        }
      }
    }
    #pragma unroll
    for (int k0 = 0; k0 < 12; k0 += 4) {   // bias GEMM: co(16x12) @ b1t^T(12x64)
      const int k = k0 + 2 * hl;
      const v2f a = *(const v2f*)&co_s[ar * COS + k];
      #pragma unroll
      for (int j = 0; j < 4; ++j) {
        const v2f b = *(const v2f*)&b1t[(j * 16 + nm16) * B1TS + k];
        acc[j] = wmma_f32(a, b, acc[j]);
      }
    }
    // write h1 back into the same rows (all h0 reads above precede these stores)
    #pragma unroll
    for (int j = 0; j < 4; ++j)
      #pragma unroll
      for (int v = 0; v < 8; ++v)
        h_s[(r0 + v + 8 * hl) * HS + j * 16 + nm16] = fmaxf(acc[j][v], 0.0f);
  }

  // ---- layer 2: sum_g (co[:,g]*h1) @ W2[g] + co @ b2 ; N=16 tile, 3 valid ----
  {
    v8f acc = {};
    const float nmask = (nm16 < COUT) ? 1.0f : 0.0f;
    const int   nc    = (nm16 < COUT) ? nm16 : 0;
    for (int g = 0; g < G; ++g) {
      const float cg = co_s[ar * COS + g];
      const float* __restrict__ Wg = W2 + g * H * COUT;
      #pragma unroll
      for (int k0 = 0; k0 < H; k0 += 4) {
        const int k = k0 + 2 * hl;
        const v2f a = (*(const v2f*)&h_s[ar * HS + k]) * cg;
        v2f b; b[0] = nmask * Wg[k * COUT + nc];
               b[1] = nmask * Wg[(k + 1) * COUT + nc];
        acc = wmma_f32(a, b, acc);
      }
    }
    #pragma unroll
    for (int k0 = 0; k0 < 12; k0 += 4) {   // bias: co(16x12) @ b2(12x3 padded)
      const int k = k0 + 2 * hl;
      const v2f a = *(const v2f*)&co_s[ar * COS + k];
      const float m0 = (k     < G) ? nmask : 0.0f;
      const float m1 = (k + 1 < G) ? nmask : 0.0f;
      const int  kk0 = (k     < G) ? k     : 0;
      const int  kk1 = (k + 1 < G) ? k + 1 : 0;
      v2f b; b[0] = m0 * b2[kk0 * COUT + nc];
             b[1] = m1 * b2[kk1 * COUT + nc];
      acc = wmma_f32(a, b, acc);
    }
    if (nm16 < COUT) {
      #pragma unroll
      for (int v = 0; v < 8; ++v)
        out[(row0 + r0 + v + 8 * hl) * COUT + nm16] = acc[v];
    }
  }
}

extern "C" void kernel_launch(void* const* d_in, const int* in_sizes, int n_in,
                              void* d_out, int out_size, void* d_ws, size_t ws_size,
                              hipStream_t stream) {
  const float* x  = (const float*)d_in[0];
  const float* co = (const float*)d_in[1];
  const float* W0 = (const float*)d_in[2];
  const float* W1 = (const float*)d_in[3];
  const float* W2 = (const float*)d_in[4];
  const float* b0 = (const float*)d_in[5];
  const float* b1 = (const float*)d_in[6];
  const float* b2 = (const float*)d_in[7];
  float* out = (float*)d_out;

  const int B = in_sizes[0] / CIN;           // 65536
  const int grid = (B + MT - 1) / MT;        // 512 workgroups
  adaptive_mlp_kernel<<<grid, 256, 0, stream>>>(x, co, W0, W1, W2, b0, b1, b2, out);

  (void)n_in; (void)out_size; (void)d_ws; (void)ws_size;
}